// GRU_867583394172
// MI455X (gfx1250) — compile-verified
//
#include <hip/hip_runtime.h>
#include <hip/hip_bf16.h>

// GRU over T=8192 steps, N=1024, D_IN=1024 on MI455X (gfx1250, wave32).
//
// Key algebraic fact (faithful to the reference's own bug):
//   h_new = (1-u)*hc + u*hc == hc   -> the u-gate (Wu, bu, hUW, hUB) is dead.
//
// Phase 1 (parallel, memory-bound ~5us at 23.3 TB/s): xT = x@Wx+bx written
//   IN PLACE into d_out (step t reads xT[t,c] before overwriting it with h),
//   xR = x@Wr+br into workspace. fp32 V_WMMA_F32_16X16X4_F32 (precision-free,
//   still far above the ~8 TFLOP/s the memory roofline demands).
//
// Phase 2 (sequential scan, latency-bound): 16 workgroups, each owning a
//   64-column slab of hRW and hW converted to bf16 in LDS (2 x 128.25 KB
//   padded slabs + staging buffers < 320 KB/WGP). Per step:
//   TDM-stage h -> matvec -> r gate -> exchange r*h via L2 -> TDM-stage r*h
//   -> matvec -> h_new. The 4 KB h / r*h vectors are staged L2->LDS with
//   tensor_load_to_lds (Tensor Data Mover, TENSORcnt) from loop-invariant D#
//   descriptors built once in SGPRs. Sync uses the CDNA5 cluster barrier
//   (s_barrier_signal/-wait -3; S_NOP when not dispatched as a cluster) plus
//   a functional sense-reversing atomic barrier for plain-grid correctness.

#define T_DIM 8192
#define DIN   1024
#define NN    1024

#define NBLK  16      // one workgroup per WGP slab; matches max cluster size
#define TPB2  512
#define KP    1026    // bf16 slab stride: dword stride 513 (odd) -> no bank conflicts

typedef float v2f __attribute__((ext_vector_type(2)));
typedef float v8f __attribute__((ext_vector_type(8)));
typedef unsigned u32x4 __attribute__((ext_vector_type(4)));
typedef unsigned u32x8 __attribute__((ext_vector_type(8)));

__device__ __forceinline__ float bits2f(unsigned u) {
  union { unsigned i; float f; } v; v.i = u; return v.f;
}
__device__ __forceinline__ unsigned short f2bf(float f) {
  union { float f; unsigned i; } v; v.f = f;
  unsigned r = v.i + 0x7fffu + ((v.i >> 16) & 1u);   // round-to-nearest-even
  return (unsigned short)(r >> 16);
}

// ---------------------------------------------------------------------------
// Tensor DMA descriptors (D#) for a contiguous 1024 x f32 tile, global -> LDS.
// Group 0: [1:0]=count=1, [63:32]=lds_addr, [120:64]=global_addr,
//          [127:126]=type=2 ("image").
// Group 1: [17:16]=data_size=2 (4 B), tensor_dim0=1024 ([79:48]),
//          tensor_dim1=1 ([111:80]), tile_dim0=1024 ([127:112]),
//          tensor_dim0_stride=1024 ([207:160]); everything else 0.
// ---------------------------------------------------------------------------
__device__ __forceinline__ u32x4 tdm_g0(unsigned lds_off, const void* gptr) {
  unsigned long long ga = (unsigned long long)(size_t)gptr;
  u32x4 g;
  g.x = 1u;                                              // count=1, user mode
  g.y = lds_off;
  g.z = (unsigned)ga;
  g.w = ((unsigned)(ga >> 32) & 0x01FFFFFFu) | 0x80000000u;  // type=2
  return g;
}
__device__ __forceinline__ u32x8 tdm_g1_vec1024() {
  u32x8 g = {};
  g[0] = 0x00020000u;            // data_size = 4 bytes
  g[1] = (1024u & 0xFFFFu) << 16;      // tensor_dim0[15:0]
  g[2] = (1024u >> 16) | (1u << 16);   // tensor_dim0[31:16], tensor_dim1=1
  g[3] = 1024u << 16;            // tile_dim0 = 1024
  g[4] = 0u;                     // tile_dim1/2 unused
  g[5] = 1024u;                  // tensor_dim0_stride[31:0]
  g[6] = 0u;
  g[7] = 0u;
  return g;
}
__device__ __forceinline__ void tdm_load_to_lds(u32x4 g0, u32x8 g1) {
  // DEV scope so the DMA observes other WGPs' stores through L2.
  asm volatile("tensor_load_to_lds %0, %1 scope:SCOPE_DEV"
               :: "s"(g0), "s"(g1) : "memory");
}

// ---------------------------------------------------------------------------
// Init: h state, barrier scalars.
// ---------------------------------------------------------------------------
__global__ void scan_init(const float* __restrict__ h0, float* h_glob,
                          int* counter, int* sense) {
  int i = threadIdx.x;
  if (i < NN) h_glob[i] = h0[i];
  if (i == 0) { *counter = 0; *sense = 0; }
}

// ---------------------------------------------------------------------------
// Phase 1: xT = x@Wx + bx (into d_out), xR = x@Wr + br (into ws).
// One wave per 16x16 C tile, 8 waves/block cover 128 rows x 16 cols.
// A (x tile) is shared between both accumulators -> halves A traffic.
// ---------------------------------------------------------------------------
__global__ __launch_bounds__(256) void gates_gemm(
    const float* __restrict__ x,
    const float* __restrict__ Wx, const float* __restrict__ bx,
    const float* __restrict__ Wr, const float* __restrict__ br,
    float* __restrict__ xT_out, float* __restrict__ xR_out) {
  const int lane  = threadIdx.x & 31;
  const int wave  = threadIdx.x >> 5;
  const int tileM = (blockIdx.y * 8 + wave) * 16;
  const int tileN = blockIdx.x * 16;
  const int mr = lane & 15;        // A row within tile == B col within tile
  const int kq = (lane >> 4) * 2;  // lanes 0-15 hold K={0,1}, lanes 16-31 K={2,3}

  v8f accT = {}; v8f accR = {};
  const float* aptr = x  + (size_t)(tileM + mr) * DIN + kq;
  const float* bxp  = Wx + (size_t)kq * NN + tileN + mr;
  const float* brp  = Wr + (size_t)kq * NN + tileN + mr;

  for (int k0 = 0; k0 < DIN; k0 += 4) {
    v2f a;  a.x  = aptr[0];   a.y  = aptr[1];
    v2f bT; bT.x = bxp[0];    bT.y = bxp[NN];
    v2f bR; bR.x = brp[0];    bR.y = brp[NN];
    accT = __builtin_amdgcn_wmma_f32_16x16x4_f32(false, a, false, bT,
                                                 (short)0, accT, false, false);
    accR = __builtin_amdgcn_wmma_f32_16x16x4_f32(false, a, false, bR,
                                                 (short)0, accR, false, false);
    aptr += 4; bxp += 4 * NN; brp += 4 * NN;
  }

  // C/D layout: VGPR i -> row tileM + i + 8*(lane>=16), col tileN + (lane&15).
  const int col   = tileN + mr;
  const int rbase = tileM + ((lane >> 4) << 3);
  const float biasT = bx[col], biasR = br[col];
#pragma unroll
  for (int i = 0; i < 8; ++i) {
    size_t off = (size_t)(rbase + i) * NN + col;
    xT_out[off] = accT[i] + biasT;
    xR_out[off] = accR[i] + biasR;
  }
}

// ---------------------------------------------------------------------------
// Cross-workgroup sync for the scan.
// Cluster barrier per ISA-recommended pattern (one wave signals, all wait);
// treated as S_NOP when ClusterID==0, so the atomic sense barrier below
// guarantees correctness under a plain grid launch as well.
// ---------------------------------------------------------------------------
__device__ __forceinline__ void scan_sync(int* counter, int* sense,
                                          int* lds_sense) {
  __syncthreads();
  if ((threadIdx.x >> 5) == 0) {
    asm volatile("s_barrier_signal -3" ::: "memory");
  }
  asm volatile("s_barrier_wait -3" ::: "memory");
  if (threadIdx.x == 0) {
    int s = *lds_sense ^ 1;
    *lds_sense = s;
    __threadfence();                         // release my stores
    if (atomicAdd(counter, 1) == NBLK - 1) {
      atomicExch(counter, 0);
      __threadfence();
      atomicExch(sense, s);
    } else {
      while (atomicAdd(sense, 0) != s) __builtin_amdgcn_s_sleep(2);
    }
    __threadfence();                         // acquire others' stores
  }
  __syncthreads();
}

// ---------------------------------------------------------------------------
// Phase 2: sequential scan. Block b owns output columns [b*64, b*64+64).
// Weights live in LDS as bf16; h / r*h exchanged through L2 and staged into
// LDS by the Tensor Data Mover. tid -> (c_local = tid&63, kseg = tid>>6).
// ---------------------------------------------------------------------------
__global__ __launch_bounds__(TPB2) void gru_scan(
    const float* __restrict__ hW, const float* __restrict__ hRW,
    const float* __restrict__ hB, const float* __restrict__ hRB,
    const float* __restrict__ xR, float* out,
    float* h_glob, float* rh_glob, int* counter, int* sense) {
  extern __shared__ unsigned char smem[];
  unsigned short* w_r  = (unsigned short*)smem;            // 64*KP bf16
  unsigned short* w_h  = w_r + 64 * KP;                    // 64*KP bf16
  float* h_lds  = (float*)(w_h + 64 * KP);                 // 1024 f32
  float* rh_lds = h_lds + NN;                              // 1024 f32
  float* red    = rh_lds + NN;                             // 8*64 f32
  __shared__ int lds_sense;

  const int tid     = threadIdx.x;
  const int blk     = blockIdx.x;
  const int c_local = tid & 63;
  const int kseg    = tid >> 6;            // 0..7
  const int gc      = blk * 64 + c_local;  // global column (threads 0..63)

  if (tid == 0) lds_sense = 0;

  // Stage this block's weight slabs into LDS as bf16 (coalesced global reads,
  // conflict-free LDS writes thanks to KP padding).
  for (int idx = tid; idx < NN * 64; idx += TPB2) {
    int k = idx >> 6, c = idx & 63;
    w_r[c * KP + k] = f2bf(hRW[(size_t)k * NN + blk * 64 + c]);
    w_h[c * KP + k] = f2bf(hW [(size_t)k * NN + blk * 64 + c]);
  }
  const float biasR = hRB[blk * 64 + (tid & 63)];
  const float biasH = hB [blk * 64 + (tid & 63)];
  __syncthreads();

  volatile float* vh  = h_glob;
  volatile float* vrh = rh_glob;
  const unsigned* wr32 = (const unsigned*)(w_r + c_local * KP) + kseg * 64;
  const unsigned* wh32 = (const unsigned*)(w_h + c_local * KP) + kseg * 64;

  // Loop-invariant TDM descriptors: h_glob -> h_lds, rh_glob -> rh_lds.
  const u32x4 g0h = tdm_g0((unsigned)(size_t)h_lds,  (const void*)h_glob);
  const u32x4 g0r = tdm_g0((unsigned)(size_t)rh_lds, (const void*)rh_glob);
  const u32x8 g1v = tdm_g1_vec1024();

  for (int t = 0; t < T_DIM; ++t) {
    // ---- TDM-stage h (L2 -> LDS), matvec 1: h @ hRW ----
    if (tid < 32) {                       // one wave issues the DMA
      tdm_load_to_lds(g0h, g1v);
      __builtin_amdgcn_s_wait_tensorcnt(0);
    }
    __syncthreads();
    {
      const float* hseg = h_lds + kseg * 128;
      float acc = 0.f;
#pragma unroll 16
      for (int kk = 0; kk < 64; ++kk) {
        unsigned w2 = wr32[kk];
        acc += hseg[2 * kk]     * bits2f(w2 << 16);
        acc += hseg[2 * kk + 1] * bits2f(w2 & 0xffff0000u);
      }
      red[kseg * 64 + c_local] = acc;
    }
    // Pull next step's streamed rows toward this WGP while we reduce.
    if (t + 1 < T_DIM && tid < 64) {
      __builtin_prefetch(&xR [(size_t)(t + 1) * NN + gc], 0, 1);
      __builtin_prefetch(&out[(size_t)(t + 1) * NN + gc], 0, 1);
    }
    __syncthreads();
    if (tid < 64) {
      float s = xR[(size_t)t * NN + gc] + biasR;
#pragma unroll
      for (int j = 0; j < 8; ++j) s += red[j * 64 + c_local];
      float r = 1.f / (1.f + __expf(-s));
      vrh[gc] = r * h_lds[gc];          // publish my 64 entries of r*h
    }
    scan_sync(counter, sense, &lds_sense);

    // ---- TDM-stage r*h, matvec 2: (r*h) @ hW ----
    if (tid < 32) {
      tdm_load_to_lds(g0r, g1v);
      __builtin_amdgcn_s_wait_tensorcnt(0);
    }
    __syncthreads();
    {
      const float* hseg = rh_lds + kseg * 128;
      float acc = 0.f;
#pragma unroll 16
      for (int kk = 0; kk < 64; ++kk) {
        unsigned w2 = wh32[kk];
        acc += hseg[2 * kk]     * bits2f(w2 << 16);
        acc += hseg[2 * kk + 1] * bits2f(w2 & 0xffff0000u);
      }
      red[kseg * 64 + c_local] = acc;
    }
    __syncthreads();
    if (tid < 64) {
      float s = out[(size_t)t * NN + gc] + biasH;   // out currently holds xT
#pragma unroll
      for (int j = 0; j < 8; ++j) s += red[j * 64 + c_local];
      float hn = tanhf(s);
      out[(size_t)t * NN + gc] = hn;    // overwrite xT[t] with h[t]
      vh[gc] = hn;                      // publish next-step state
    }
    scan_sync(counter, sense, &lds_sense);
  }
}

// ---------------------------------------------------------------------------
// Host entry.  ws layout (floats): [0,1024) h, [1024,2048) r*h,
// [2048] counter, [2049] sense, [4096, 4096+T*N) xR.  Needs ~32.02 MB ws.
// ---------------------------------------------------------------------------
extern "C" void kernel_launch(void* const* d_in, const int* in_sizes, int n_in,
                              void* d_out, int out_size, void* d_ws,
                              size_t ws_size, hipStream_t stream) {
  const float* x   = (const float*)d_in[0];
  const float* Wx  = (const float*)d_in[1];
  const float* bx  = (const float*)d_in[2];
  // d_in[3]=Wu, d_in[4]=bu, d_in[8]=hUW, d_in[11]=hUB: dead (u-gate cancels)
  const float* Wr  = (const float*)d_in[5];
  const float* br  = (const float*)d_in[6];
  const float* hW  = (const float*)d_in[7];
  const float* hRW = (const float*)d_in[9];
  const float* hB  = (const float*)d_in[10];
  const float* hRB = (const float*)d_in[12];
  const float* h0  = (const float*)d_in[13];

  float* out     = (float*)d_out;
  float* ws      = (float*)d_ws;
  float* h_glob  = ws;
  float* rh_glob = ws + 1024;
  int*   counter = (int*)(ws + 2048);
  int*   sense   = counter + 1;
  float* xR      = ws + 4096;

  scan_init<<<1, 1024, 0, stream>>>(h0, h_glob, counter, sense);

  dim3 g1(NN / 16, T_DIM / 128);
  gates_gemm<<<g1, 256, 0, stream>>>(x, Wx, bx, Wr, br, out, xR);

  const size_t smem_bytes = (size_t)2 * 64 * KP * sizeof(unsigned short) +
                            (2 * NN + 8 * 64) * sizeof(float);  // 272,896 B
  hipFuncSetAttribute(reinterpret_cast<const void*>(gru_scan),
                      hipFuncAttributeMaxDynamicSharedMemorySize,
                      (int)smem_bytes);   // idempotent, called every launch
  gru_scan<<<NBLK, TPB2, smem_bytes, stream>>>(hW, hRW, hB, hRB, xR, out,
                                               h_glob, rh_glob, counter, sense);
}